// AgentOneHotEncoder_21354577396017
// MI455X (gfx1250) — compile-verified
//
#include <hip/hip_runtime.h>
#include <stdint.h>

// Problem constants (from reference):
#define DEPTH   64
#define OUT     256
#define TOKENS  (64 * 8192)   // B*T = 524288

#define THREADS 256
#define BLOCKS  1024
#define WPB     (THREADS / 32)  // waves per block (wave32)

typedef float        v4f __attribute__((ext_vector_type(4)));
typedef unsigned int v4u __attribute__((ext_vector_type(4)));
typedef int          v8i __attribute__((ext_vector_type(8)));
typedef int          v4i __attribute__((ext_vector_type(4)));

__global__ __launch_bounds__(THREADS)
void onehot_embed_kernel(const int* __restrict__ ids,
                         const float* __restrict__ W,     // [OUT][DEPTH] row-major
                         const float* __restrict__ bias,  // [OUT]
                         float* __restrict__ out)         // [TOKENS][OUT]
{
    extern __shared__ float lds[];
    float* wStage = lds;               // [OUT][DEPTH]  64 KB staging (TDM target)
    float* table  = lds + OUT * DEPTH; // [DEPTH][OUT]  64 KB bias-fused gather table

    const int tid  = threadIdx.x;
    const int lane = tid & 31;

    // ---- Stage W into LDS via Tensor Data Mover ----------------------------
#if __has_builtin(__builtin_amdgcn_tensor_load_to_lds)
    if (tid < 32) {  // one wave issues the TDM descriptor; EXEC is ignored by TDM
        const uint64_t ga     = (uint64_t)(const void*)W;
        const uint32_t ldsOff = (uint32_t)(uintptr_t)(void*)wStage;

        // D# group 0: count=1 | lds_addr | global_addr[56:0] | type=2
        v4u g0;
        g0[0] = 1u;                                   // count=1, user mode
        g0[1] = ldsOff;                               // lds_addr (bytes)
        g0[2] = (uint32_t)ga;                         // global_addr[31:0]
        g0[3] = (uint32_t)((ga >> 32) & 0x1FFFFFFu)   // global_addr[56:32]
              | (2u << 30);                           // type = 2 ("image")

        // D# group 1: 2D tensor, f32 elements, tile == whole tensor
        v8i g1;
        g1[0] = (int)(2u << 16);                 // wg_mask=0, data_size=2 (4B)
        g1[1] = (int)((uint32_t)DEPTH << 16);    // tensor_dim0[15:0]=64 @bits79:48
        g1[2] = (int)((uint32_t)OUT   << 16);    // tensor_dim1[15:0]=256 @bits111:80
        g1[3] = (int)((uint32_t)DEPTH << 16);    // tile_dim0=64 @bits127:112
        g1[4] = (int)OUT;                        // tile_dim1=256, tile_dim2=0
        g1[5] = (int)DEPTH;                      // tensor_dim0_stride=64 (low 32)
        g1[6] = 0;                               // stride hi / dim1_stride (unused, 2D)
        g1[7] = 0;

        v4i g2 = (v4i){0, 0, 0, 0};              // 2D: groups 2/3 unused
        v4i g3 = (v4i){0, 0, 0, 0};
        v8i gx = (v8i){0, 0, 0, 0, 0, 0, 0, 0};  // extra group (clang-23 6-arg form)

        __builtin_amdgcn_tensor_load_to_lds(g0, g1, g2, g3, gx, 0);
        __builtin_amdgcn_s_wait_tensorcnt(0);
    }
#else
    // Fallback: cooperative copy of W into LDS
    for (int i = tid; i < OUT * DEPTH; i += THREADS)
        wStage[i] = W[i];
#endif
    __syncthreads();

    // ---- Build bias-fused transposed table: table[d][o] = W[o][d] + b[o] ---
    {
        const float bv = bias[tid];              // THREADS == OUT == 256
#pragma unroll 8
        for (int k = 0; k < DEPTH; ++k)
            table[k * OUT + tid] = wStage[tid * DEPTH + k] + bv;
    }
    __syncthreads();

    // ---- Streaming gather: each wave emits 32 tokens per pass --------------
    const int  waveId = blockIdx.x * WPB + (tid >> 5);
    const long nWaves = (long)gridDim.x * WPB;
    const long stride = nWaves * 32;

    for (long base = (long)waveId * 32; base < TOKENS; base += stride) {
        const int myId = ids[base + lane];             // coalesced 32 ids
        __builtin_prefetch(ids + base + stride, 0, 3); // global_prefetch_b8

#pragma unroll
        for (int j = 0; j < 32; ++j) {
            const int id = __builtin_amdgcn_readlane(myId, j);  // SGPR id
            const v4f* row = (const v4f*)(table + id * OUT);    // 1 KB LDS row
            const v4f lo = row[lane];                           // ds_load_b128
            const v4f hi = row[lane + 32];                      // ds_load_b128
            v4f* o = (v4f*)(out + (base + j) * (long)OUT);
            __builtin_nontemporal_store(lo, o + lane);          // b128 store, TH=NT
            __builtin_nontemporal_store(hi, o + lane + 32);     // b128 store, TH=NT
        }
    }
}

extern "C" void kernel_launch(void* const* d_in, const int* in_sizes, int n_in,
                              void* d_out, int out_size, void* d_ws, size_t ws_size,
                              hipStream_t stream) {
    const int*   ids = (const int*)d_in[0];    // [B,T,1] integer ids
    const float* W   = (const float*)d_in[1];  // [OUT, DEPTH] f32
    const float* b   = (const float*)d_in[2];  // [OUT] f32
    float*       out = (float*)d_out;          // [B,T,1,OUT] f32

    const size_t shmem = (size_t)2 * OUT * DEPTH * sizeof(float); // 128 KB LDS
    onehot_embed_kernel<<<BLOCKS, THREADS, shmem, stream>>>(ids, W, b, out);
}